// LinearAttentionModel_55800215109664
// MI455X (gfx1250) — compile-verified
//
#include <hip/hip_runtime.h>
#include <hip/hip_bf16.h>

#define DP1 2049   // d+1
#define LP1 8193   // l+1
#define NTILES 513 // ceil(8193/16) column tiles of 16
#define KCH 17     // ceil(2049/128) i-chunks for K^T matvec

typedef __attribute__((ext_vector_type(2))) float v2f;
typedef __attribute__((ext_vector_type(8))) float v8f;

// Workspace layout (floats)
#define WS_ZLAST 0
#define WS_Q     (WS_ZLAST + DP1)        // 2049
#define WS_W2    (WS_Q + DP1)            // 4098  (k/vrow interleaved, 2*DP1)
#define WS_KPART (WS_W2 + 2 * DP1)       // 8196  (KCH * DP1)
#define WS_WPART (WS_KPART + KCH * DP1)  // 43029 (NTILES)

// ---------------------------------------------------------------------------
// Kernel 0: extract last column of Z (strided gather, tiny)
__global__ void lam_extract_zlast(const float* __restrict__ Z, float* __restrict__ zlast) {
  int c = blockIdx.x * 256 + threadIdx.x;
  if (c < DP1) zlast[c] = Z[(size_t)c * LP1 + (LP1 - 1)];
}

// ---------------------------------------------------------------------------
// Kernel 1: q = Q @ zlast. One block per row, coalesced row read, tree reduce.
__global__ __launch_bounds__(256) void lam_qmatvec(const float* __restrict__ Q,
                                                   const float* __restrict__ zlast,
                                                   float* __restrict__ q) {
  const int r = blockIdx.x;
  float acc = 0.f;
  for (int c = threadIdx.x; c < DP1; c += 256)
    acc += Q[(size_t)r * DP1 + c] * zlast[c];
  for (int off = 16; off; off >>= 1) acc += __shfl_xor(acc, off, 32);
  __shared__ float red[8];
  if ((threadIdx.x & 31) == 0) red[threadIdx.x >> 5] = acc;
  __syncthreads();
  if (threadIdx.x == 0) {
    float t = 0.f;
    for (int w = 0; w < 8; ++w) t += red[w];
    q[r] = t;
  }
}

// ---------------------------------------------------------------------------
// Kernel 2a: kpart[ch][c] = sum_{i in chunk} K[i][c] * q[i]
__global__ __launch_bounds__(256) void lam_ktpart(const float* __restrict__ K,
                                                  const float* __restrict__ q,
                                                  float* __restrict__ kpart) {
  const int c = blockIdx.x * 256 + threadIdx.x;
  if (c >= DP1) return;
  const int i0 = blockIdx.y * 128;
  const int i1 = (i0 + 128 < DP1) ? (i0 + 128) : DP1;
  float acc = 0.f;
  for (int i = i0; i < i1; ++i)
    acc += K[(size_t)i * DP1 + c] * q[i];
  kpart[(size_t)blockIdx.y * DP1 + c] = acc;
}

// Kernel 2b: W2[2c] = k[c] = sum over chunks; W2[2c+1] = V[-1][c]
__global__ void lam_w2build(const float* __restrict__ kpart,
                            const float* __restrict__ V,
                            float* __restrict__ W2) {
  const int c = blockIdx.x * 256 + threadIdx.x;
  if (c >= DP1) return;
  float acc = 0.f;
  for (int ch = 0; ch < KCH; ++ch) acc += kpart[(size_t)ch * DP1 + c];
  W2[2 * c]     = acc;
  W2[2 * c + 1] = V[(size_t)(DP1 - 1) * DP1 + c];
}

// ---------------------------------------------------------------------------
// Kernel 3 (dominant, streams Z once): 4 waves per 16-column tile, each wave
// covers a quarter of the i-range; partial D accumulators are additive, so
// they are combined in LDS before the s*v pairing.
// WMMA f32 16x16x4:  D(16x16) += A(16x4) x B(4x16)
//   A[m,kk] = Z[i0+kk, j0+m]; B[kk,0]=k[i0+kk], B[kk,1]=vrow[i0+kk], else 0.
// Main loop is branch-free: clamped addresses + v_cndmask zeroing.
__global__ __launch_bounds__(256) void lam_fused_zstream(const float* __restrict__ Z,
                                                         const float* __restrict__ W2,
                                                         float* __restrict__ wpart) {
  const int lane  = threadIdx.x & 31;
  const int wave  = threadIdx.x >> 5;
  const int grp   = wave >> 2;              // 2 tile-groups per block
  const int chunk = wave & 3;               // i-range quarter
  int tile = blockIdx.x * 2 + grp;
  const bool owner = (tile < NTILES);
  if (!owner) tile = NTILES - 1;            // clamp: duplicate work, no store
  const int j0 = tile * 16;
  const int h  = lane >> 4;                 // lane half -> K rows {2h, 2h+1}
  const int m  = lane & 15;                 // A row M (output column index)
  int jc = j0 + m;
  const bool jvalid = (jc < LP1);
  if (!jvalid) jc = LP1 - 1;                // clamp column address
  const bool bmask = (m < 2);
  const int msel = bmask ? m : 0;

  const int iBeg  = (chunk * DP1) >> 2;       // 0,512,1024,1536
  const int iEnd  = ((chunk + 1) * DP1) >> 2; // 512,1024,1536,2049
  const int nFull = (iEnd - iBeg) >> 2;       // 128 full K-groups each

  const float* pa = Z + (size_t)(iBeg + 2 * h) * LP1 + jc;
  const float* pb = W2 + 2 * (iBeg + 2 * h) + msel;

  v8f acc = {0.f, 0.f, 0.f, 0.f, 0.f, 0.f, 0.f, 0.f};
#pragma unroll 2
  for (int it = 0; it < nFull; ++it) {
    float a0 = pa[0];
    float a1 = pa[LP1];
    __builtin_prefetch(pa + 32 * (size_t)LP1, 0, 0);  // stream ahead into GL2
    float b0 = pb[0];
    float b1 = pb[2];
    a0 = jvalid ? a0 : 0.f;
    a1 = jvalid ? a1 : 0.f;
    b0 = bmask ? b0 : 0.f;
    b1 = bmask ? b1 : 0.f;
    v2f a; a[0] = a0; a[1] = a1;
    v2f b; b[0] = b0; b[1] = b1;
    acc = __builtin_amdgcn_wmma_f32_16x16x4_f32(false, a, false, b, (short)0,
                                                acc, false, false);
    pa += 4 * (size_t)LP1;
    pb += 8;
  }
  // Row tail (only last chunk has 1 leftover row): one predicated WMMA.
  const int it0 = iBeg + (nFull << 2);
  if (it0 < iEnd) {
    const int r0 = it0 + 2 * h;
    const int r1 = r0 + 1;
    float a0 = (jvalid && r0 < iEnd) ? Z[(size_t)r0 * LP1 + jc] : 0.f;
    float a1 = (jvalid && r1 < iEnd) ? Z[(size_t)r1 * LP1 + jc] : 0.f;
    float b0 = (bmask && r0 < iEnd) ? W2[2 * r0 + msel] : 0.f;
    float b1 = (bmask && r1 < iEnd) ? W2[2 * r1 + msel] : 0.f;
    v2f a; a[0] = a0; a[1] = a1;
    v2f b; b[0] = b0; b[1] = b1;
    acc = __builtin_amdgcn_wmma_f32_16x16x4_f32(false, a, false, b, (short)0,
                                                acc, false, false);
  }

  // Combine the 4 chunk-waves' partial D matrices via LDS (lane-wise add).
  __shared__ float red[8][8][32];  // [wave][vgpr][lane] = 8 KB
#pragma unroll
  for (int r = 0; r < 8; ++r) red[wave][r][lane] = acc[r];
  __syncthreads();
  if (chunk == 0) {
    // D layout: lane N=0 holds s_{j0+r(+8 if lane>=16)} in slot r; N=1 holds v.
    float partial = 0.f;
#pragma unroll
    for (int r = 0; r < 8; ++r) {
      float s = red[wave][r][lane] + red[wave + 1][r][lane] +
                red[wave + 2][r][lane] + red[wave + 3][r][lane];
      float o = __shfl_xor(s, 1, 32);
      partial += s * o;
    }
    float tot = __shfl(partial, 0, 32) + __shfl(partial, 16, 32);
    if (lane == 0 && owner) wpart[tile] = tot;  // one writer: deterministic
  }
}

// ---------------------------------------------------------------------------
// Kernel 4: final fixed-order reduction of per-tile partials, scale by 1/8192
__global__ __launch_bounds__(256) void lam_final(const float* __restrict__ wpart,
                                                 float* __restrict__ out) {
  float acc = 0.f;
  for (int i = threadIdx.x; i < NTILES; i += 256) acc += wpart[i];
  for (int off = 16; off; off >>= 1) acc += __shfl_xor(acc, off, 32);
  __shared__ float red[8];
  if ((threadIdx.x & 31) == 0) red[threadIdx.x >> 5] = acc;
  __syncthreads();
  if (threadIdx.x == 0) {
    float t = 0.f;
    for (int w = 0; w < 8; ++w) t += red[w];
    out[0] = t * (1.0f / 8192.0f);
  }
}

// ---------------------------------------------------------------------------
extern "C" void kernel_launch(void* const* d_in, const int* in_sizes, int n_in,
                              void* d_out, int out_size, void* d_ws, size_t ws_size,
                              hipStream_t stream) {
  (void)in_sizes; (void)n_in; (void)out_size; (void)ws_size;
  const float* Z = (const float*)d_in[0];
  const float* V = (const float*)d_in[1];
  const float* K = (const float*)d_in[2];
  const float* Q = (const float*)d_in[3];
  float* ws    = (float*)d_ws;
  float* zlast = ws + WS_ZLAST;
  float* q     = ws + WS_Q;
  float* W2    = ws + WS_W2;
  float* kpart = ws + WS_KPART;
  float* wpart = ws + WS_WPART;
  float* out   = (float*)d_out;

  lam_extract_zlast<<<(DP1 + 255) / 256, 256, 0, stream>>>(Z, zlast);
  lam_qmatvec<<<DP1, 256, 0, stream>>>(Q, zlast, q);
  dim3 g2((DP1 + 255) / 256, KCH);
  lam_ktpart<<<g2, 256, 0, stream>>>(K, q, kpart);
  lam_w2build<<<(DP1 + 255) / 256, 256, 0, stream>>>(kpart, V, W2);
  lam_fused_zstream<<<(NTILES + 1) / 2, 256, 0, stream>>>(Z, W2, wpart);
  lam_final<<<1, 256, 0, stream>>>(wpart, out);
}